// GATIIN_72756745994562
// MI455X (gfx1250) — compile-verified
//
#include <hip/hip_runtime.h>
#include <hip/hip_bf16.h>

typedef float v2f __attribute__((ext_vector_type(2)));
typedef float v8f __attribute__((ext_vector_type(8)));

#define HID 256
#define HEADS 8
#define DHEAD 32

// ---------------------------------------------------------------------------
// Generic f32 WMMA GEMM:  C[M,Nc] = A[M,K] @ B[K,Nc] (+ bias[Nc] if non-null)
// One wave -> 16x64 output strip (four 16x16 WMMA tiles sharing the A fragment).
// Uses V_WMMA_F32_16X16X4_F32 (fp32 in, fp32 accumulate).  Nc must be %64.
// ---------------------------------------------------------------------------
__global__ __launch_bounds__(128)
void gatiin_gemm_wmma_f32(const float* __restrict__ A, const float* __restrict__ B,
                          const float* __restrict__ bias, float* __restrict__ C,
                          int M, int Nc, int K)
{
    const int wave = blockIdx.x * (blockDim.x >> 5) + (threadIdx.x >> 5);
    const int lane = threadIdx.x & 31;
    const int tilesN4 = Nc >> 6;              // groups of four 16-wide tiles
    const int tilesM  = (M + 15) >> 4;
    if (wave >= tilesM * tilesN4) return;     // uniform per-wave exit (EXEC stays all-1s)

    const int tm = wave / tilesN4;
    const int tn = wave % tilesN4;

    int rowA = tm * 16 + (lane & 15);
    if (rowA >= M) rowA = M - 1;              // clamp; garbage rows masked at store
    const int kHalf = (lane >> 4) << 1;       // 0 or 2
    const int colB  = tn * 64 + (lane & 15);

    const float* __restrict__ Arow = A + (size_t)rowA * K;
    v8f c0 = {}; v8f c1 = {}; v8f c2 = {}; v8f c3 = {};

    for (int k = 0; k < K; k += 4) {
        v2f a;                                 // A 16x4: lane=M, VGPR0=K+kHalf, VGPR1=K+kHalf+1
        a.x = Arow[k + kHalf];
        a.y = Arow[k + kHalf + 1];
        const float* __restrict__ Bk = B + (size_t)(k + kHalf) * Nc;
        v2f b0, b1, b2, b3;                    // B 4x16: lane=N, upper half-wave = rows +2
        b0.x = Bk[colB];       b0.y = Bk[Nc + colB];
        b1.x = Bk[colB + 16];  b1.y = Bk[Nc + colB + 16];
        b2.x = Bk[colB + 32];  b2.y = Bk[Nc + colB + 32];
        b3.x = Bk[colB + 48];  b3.y = Bk[Nc + colB + 48];
        c0 = __builtin_amdgcn_wmma_f32_16x16x4_f32(false, a, false, b0, (short)0, c0, false, false);
        c1 = __builtin_amdgcn_wmma_f32_16x16x4_f32(false, a, false, b1, (short)0, c1, false, false);
        c2 = __builtin_amdgcn_wmma_f32_16x16x4_f32(false, a, false, b2, (short)0, c2, false, false);
        c3 = __builtin_amdgcn_wmma_f32_16x16x4_f32(false, a, false, b3, (short)0, c3, false, false);
    }

    // C/D layout: VGPR r holds M = r + 8*(lane>=16), N = lane&15 within the tile
    const int rBase = tm * 16 + ((lane >> 4) << 3);
    const int col0  = tn * 64 + (lane & 15);
    const float bv0 = bias ? bias[col0]      : 0.0f;
    const float bv1 = bias ? bias[col0 + 16] : 0.0f;
    const float bv2 = bias ? bias[col0 + 32] : 0.0f;
    const float bv3 = bias ? bias[col0 + 48] : 0.0f;
#pragma unroll
    for (int r = 0; r < 8; ++r) {
        const int row = rBase + r;
        if (row < M) {
            float* __restrict__ Crow = C + (size_t)row * Nc;
            Crow[col0]      = c0[r] + bv0;
            Crow[col0 + 16] = c1[r] + bv1;
            Crow[col0 + 32] = c2[r] + bv2;
            Crow[col0 + 48] = c3[r] + bv3;
        }
    }
}

// ---------------------------------------------------------------------------
// Utility fill
// ---------------------------------------------------------------------------
__global__ void gatiin_fill_f32(float* __restrict__ p, float v, long long n)
{
    long long i = (long long)blockIdx.x * blockDim.x + threadIdx.x;
    if (i < n) p[i] = v;
}

// ---------------------------------------------------------------------------
// GraphNorm: per-channel sum / sumsq partials (coalesced: thread = channel)
// ---------------------------------------------------------------------------
__global__ __launch_bounds__(HID)
void gatiin_colstats(const float* __restrict__ X, float* __restrict__ sum,
                     float* __restrict__ sumsq, int M)
{
    const int c  = threadIdx.x;
    const int r0 = blockIdx.x * 128;
    int rend = r0 + 128; if (rend > M) rend = M;
    float s = 0.f, q = 0.f;
    for (int r = r0; r < rend; ++r) {
        float v = X[(size_t)r * HID + c];
        s += v; q += v * v;
    }
    atomicAdd(&sum[c], s);
    atomicAdd(&sumsq[c], q);
}

__global__ void gatiin_norm_finalize(const float* __restrict__ sum, const float* __restrict__ sumsq,
                                     const float* __restrict__ w, const float* __restrict__ ms,
                                     float* __restrict__ meanms, float* __restrict__ scale, int M)
{
    const int c = threadIdx.x;
    const float mean = sum[c] / (float)M;
    const float ex2  = sumsq[c] / (float)M;
    const float m    = ms[c];
    // var of (x - ms*mean): E[x^2] - 2*ms*mean*E[x] + ms^2*mean^2
    const float var = ex2 - 2.f * m * mean * mean + m * m * mean * mean;
    meanms[c] = m * mean;
    scale[c]  = w[c] * rsqrtf(var + 1e-5f);
}

__global__ void gatiin_norm_apply(const float* __restrict__ X, const float* __restrict__ meanms,
                                  const float* __restrict__ scale, const float* __restrict__ b,
                                  float* __restrict__ Y, int total)
{
    int i = blockIdx.x * blockDim.x + threadIdx.x;
    if (i >= total) return;
    int c = i & (HID - 1);
    Y[i] = (X[i] - meanms[c]) * scale[c] + b[c];
}

// ---------------------------------------------------------------------------
// Edge phase
// ---------------------------------------------------------------------------
__device__ __forceinline__ void atomicMaxF(float* addr, float val)
{
    unsigned int* ua = (unsigned int*)addr;
    unsigned int old = *ua;
    while (__uint_as_float(old) < val) {
        unsigned int assumed = old;
        old = atomicCAS(ua, assumed, __float_as_uint(val));
        if (old == assumed) break;
    }
}

__device__ __forceinline__ float4 lrelu4(float4 v, float s)
{
    v.x = v.x > 0.f ? v.x : s * v.x;
    v.y = v.y > 0.f ? v.y : s * v.y;
    v.z = v.z > 0.f ? v.z : s * v.z;
    v.w = v.w > 0.f ? v.w : s * v.w;
    return v;
}

// thread = (edge, head); 8 lanes of a wave cover one edge's full 256-f row (coalesced)
__global__ __launch_bounds__(256)
void gatiin_edge_logits(const float* __restrict__ xl, const float* __restrict__ xr,
                        const float* __restrict__ tproj, const float* __restrict__ att,
                        const int* __restrict__ src, const int* __restrict__ dst,
                        const int* __restrict__ ecode,
                        float* __restrict__ logits, float* __restrict__ segmax, int E)
{
    int idx = blockIdx.x * blockDim.x + threadIdx.x;
    if (idx >= E * HEADS) return;
    const int e = idx >> 3, h = idx & 7;
    const int s = src[e], d = dst[e], c = ecode[e];
    const float4* __restrict__ pl = (const float4*)(xl    + (size_t)s * HID + h * DHEAD);
    const float4* __restrict__ pr = (const float4*)(xr    + (size_t)d * HID + h * DHEAD);
    const float4* __restrict__ pe = (const float4*)(tproj + (size_t)c * HID + h * DHEAD);
    const float4* __restrict__ pa = (const float4*)(att   + h * DHEAD);
    float acc = 0.f;
#pragma unroll
    for (int i = 0; i < 8; ++i) {
        float4 l = pl[i], r = pr[i], ee = pe[i], a = pa[i];
        float4 m = make_float4(l.x + r.x + ee.x, l.y + r.y + ee.y,
                               l.z + r.z + ee.z, l.w + r.w + ee.w);
        m = lrelu4(m, 0.2f);
        acc += m.x * a.x + m.y * a.y + m.z * a.z + m.w * a.w;
    }
    logits[idx] = acc;
    atomicMaxF(&segmax[d * HEADS + h], acc);
}

__global__ __launch_bounds__(256)
void gatiin_edge_expsum(float* __restrict__ logits, const float* __restrict__ segmax,
                        float* __restrict__ denom, const int* __restrict__ dst, int E)
{
    int idx = blockIdx.x * blockDim.x + threadIdx.x;
    if (idx >= E * HEADS) return;
    const int e = idx >> 3, h = idx & 7;
    const int d = dst[e];
    float ex = expf(logits[idx] - segmax[d * HEADS + h]);
    logits[idx] = ex;                     // reuse buffer: now holds exp()
    atomicAdd(&denom[d * HEADS + h], ex);
}

__global__ __launch_bounds__(256)
void gatiin_edge_message(const float* __restrict__ xl, const float* __restrict__ ex,
                         const float* __restrict__ denom, const float* __restrict__ ew,
                         const int* __restrict__ src, const int* __restrict__ dst,
                         float* __restrict__ acc, int E)
{
    int idx = blockIdx.x * blockDim.x + threadIdx.x;
    if (idx >= E * HEADS) return;
    const int e = idx >> 3, h = idx & 7;
    const int s = src[e], d = dst[e];
    const float alpha = ex[idx] / (denom[d * HEADS + h] + 1e-16f);
    const float coef  = alpha * ew[e];
    const float4* __restrict__ pl = (const float4*)(xl + (size_t)s * HID + h * DHEAD);
    float* __restrict__ pa = acc + (size_t)d * HID + h * DHEAD;
#pragma unroll
    for (int i = 0; i < 8; ++i) {
        float4 v = pl[i];
        atomicAdd(pa + 4 * i + 0, v.x * coef);
        atomicAdd(pa + 4 * i + 1, v.y * coef);
        atomicAdd(pa + 4 * i + 2, v.z * coef);
        atomicAdd(pa + 4 * i + 3, v.w * coef);
    }
}

// h = ALPHA*x0 + (1-ALPHA)*leaky_relu(acc + gat_b, 0.01)
__global__ void gatiin_gat_finalize(const float* __restrict__ x0, const float* __restrict__ acc,
                                    const float* __restrict__ gb, float* __restrict__ h, int total)
{
    int i = blockIdx.x * blockDim.x + threadIdx.x;
    if (i >= total) return;
    int c = i & (HID - 1);
    float v = acc[i] + gb[c];
    v = v > 0.f ? v : 0.01f * v;
    h[i] = 0.2f * x0[i] + 0.8f * v;
}

// ---------------------------------------------------------------------------
// Host orchestration
// ---------------------------------------------------------------------------
extern "C" void kernel_launch(void* const* d_in, const int* in_sizes, int n_in,
                              void* d_out, int out_size, void* d_ws, size_t ws_size,
                              hipStream_t stream)
{
    const int F_IN = 128, OUT = 128, BLOCKS = 2, ED = 32;
    const int N  = in_sizes[0] / F_IN;      // 50000
    const int E  = in_sizes[1] / 2;         // 500000
    const int EV = in_sizes[4] / ED;        // 500

    const float* x        = (const float*)d_in[0];
    const int*   eidx     = (const int*)d_in[1];
    const float* ew       = (const float*)d_in[2];
    const int*   ecode    = (const int*)d_in[3];
    const float* etab     = (const float*)d_in[4];
    const float* pre_W    = (const float*)d_in[5];
    const float* pre_b    = (const float*)d_in[6];
    const float* pre_nw   = (const float*)d_in[7];
    const float* pre_nb   = (const float*)d_in[8];
    const float* pre_nms  = (const float*)d_in[9];
    const float* norm_w   = (const float*)d_in[10];
    const float* norm_b   = (const float*)d_in[11];
    const float* norm_ms  = (const float*)d_in[12];
    const float* Wl       = (const float*)d_in[13];
    const float* bl       = (const float*)d_in[14];
    const float* Wr       = (const float*)d_in[15];
    const float* br       = (const float*)d_in[16];
    const float* att      = (const float*)d_in[17];
    const float* We       = (const float*)d_in[18];
    const float* gat_b    = (const float*)d_in[19];
    const float* post_W   = (const float*)d_in[20];
    const float* post_b   = (const float*)d_in[21];

    const int* src = eidx;
    const int* dst = eidx + E;

    // --- workspace carve (256B aligned) ---
    char* wp = (char*)d_ws;
    auto carve = [&](size_t bytes) -> float* {
        float* p = (float*)wp;
        wp += (bytes + 255) & ~(size_t)255;
        return p;
    };
    float* x0     = carve((size_t)N * HID * 4);   // normalized input (residual anchor)
    float* hb     = carve((size_t)N * HID * 4);   // running h
    float* t0     = carve((size_t)N * HID * 4);   // hn, then message accumulator
    float* xl     = carve((size_t)N * HID * 4);
    float* xr     = carve((size_t)N * HID * 4);
    float* logits = carve((size_t)E * HEADS * 4);
    float* segmax = carve((size_t)N * HEADS * 4);
    float* denom  = carve((size_t)N * HEADS * 4);
    float* tproj  = carve((size_t)EV * HID * 4);
    float* stats  = carve(4 * HID * 4);           // sum | sumsq | meanms | scale
    float* s_sum = stats, *s_sq = stats + HID, *s_mm = stats + 2 * HID, *s_sc = stats + 3 * HID;

    auto gemm = [&](const float* A, const float* B, const float* bias, float* C,
                    int M, int Nc, int K) {
        long long waves = (long long)((M + 15) / 16) * (Nc / 64);
        int blocks = (int)((waves + 3) / 4);
        gatiin_gemm_wmma_f32<<<blocks, 128, 0, stream>>>(A, B, bias, C, M, Nc, K);
    };

    auto gnorm = [&](const float* X, const float* wv, const float* bv, const float* msv,
                     float* Y) {
        gatiin_fill_f32<<<2, 256, 0, stream>>>(s_sum, 0.f, 2 * HID);
        gatiin_colstats<<<(N + 127) / 128, HID, 0, stream>>>(X, s_sum, s_sq, N);
        gatiin_norm_finalize<<<1, HID, 0, stream>>>(s_sum, s_sq, wv, msv, s_mm, s_sc, N);
        gatiin_norm_apply<<<(N * HID + 255) / 256, 256, 0, stream>>>(X, s_mm, s_sc, bv, Y, N * HID);
    };

    const int EH = E * HEADS;
    const int ehBlocks = (EH + 255) / 256;

    // ---- pre: x0 = graph_norm(x @ pre_W + pre_b) ----
    gemm(x, pre_W, pre_b, t0, N, HID, F_IN);
    gnorm(t0, pre_nw, pre_nb, pre_nms, x0);

    const float* hcur = x0;
    for (int i = 0; i < BLOCKS; ++i) {
        // hn = graph_norm(h) -> t0
        gnorm(hcur, norm_w + i * HID, norm_b + i * HID, norm_ms + i * HID, t0);
        // xl / xr projections
        gemm(t0, Wl + (size_t)i * HID * HID, bl + i * HID, xl, N, HID, HID);
        gemm(t0, Wr + (size_t)i * HID * HID, br + i * HID, xr, N, HID, HID);
        // edge-embedding projection factored through the (small) table
        gemm(etab, We + (size_t)i * ED * HID, nullptr, tproj, EV, HID, ED);
        // segment softmax state + message accumulator (t0 is free now)
        gatiin_fill_f32<<<(N * HEADS + 255) / 256, 256, 0, stream>>>(segmax, -3.4e38f, (long long)N * HEADS);
        gatiin_fill_f32<<<(N * HEADS + 255) / 256, 256, 0, stream>>>(denom, 0.f, (long long)N * HEADS);
        gatiin_fill_f32<<<(N * HID + 255) / 256, 256, 0, stream>>>(t0, 0.f, (long long)N * HID);
        gatiin_edge_logits<<<ehBlocks, 256, 0, stream>>>(xl, xr, tproj, att + i * HEADS * DHEAD,
                                                         src, dst, ecode, logits, segmax, E);
        gatiin_edge_expsum<<<ehBlocks, 256, 0, stream>>>(logits, segmax, denom, dst, E);
        gatiin_edge_message<<<ehBlocks, 256, 0, stream>>>(xl, logits, denom, ew, src, dst, t0, E);
        gatiin_gat_finalize<<<(N * HID + 255) / 256, 256, 0, stream>>>(x0, t0, gat_b + i * HID, hb, N * HID);
        hcur = hb;
    }

    // ---- post: out = h @ post_W + post_b ----
    gemm(hcur, post_W, post_b, (float*)d_out, N, OUT, HID);
}